// SpinGlassLandscape_90460601188757
// MI455X (gfx1250) — compile-verified
//
#include <hip/hip_runtime.h>

// Problem constants (match reference)
#define NS   512                 // N_SITES
#define NQ   21                  // N_STATES
#define NB   256                 // BATCH
#define PAIR (NQ * NQ)           // 441 floats per (i,j) block
#define JC   16                  // j-blocks per LDS tile
#define CHUNK_F (JC * PAIR)      // 7056 floats = 28224 B (16B aligned stride)
#define SPLIT 2                  // j-range split per site for more blocks
#define JPART (NS / SPLIT)       // 256 j per workgroup
#define NCHUNK (JPART / JC)      // 16 tiles per workgroup

#if defined(__gfx1250__) && __has_builtin(__builtin_amdgcn_global_load_async_to_lds_b128) && __has_builtin(__builtin_amdgcn_s_wait_asynccnt)
#define USE_ASYNC 1
#else
#define USE_ASYNC 0
#endif

// Exact pointee type the builtin expects: GCC-vector of 4 ints,
// in global (AS1) / LDS (AS3) address spaces.
typedef int v4i __attribute__((vector_size(4 * sizeof(int))));
typedef __attribute__((address_space(1))) v4i as1_v4i;
typedef __attribute__((address_space(3))) v4i as3_v4i;

// Cooperatively copy CHUNK_F floats (16B-aligned) global -> LDS.
__device__ __forceinline__ void copy_chunk(const float* __restrict__ gbase,
                                           float* __restrict__ lbase, int tid) {
#if USE_ASYNC
  for (int v = tid; v < CHUNK_F / 4; v += NB) {
    float* g = const_cast<float*>(gbase) + v * 4;
    __builtin_amdgcn_global_load_async_to_lds_b128(
        (as1_v4i*)g, (as3_v4i*)(lbase + v * 4), 0, 0);
  }
#else
  for (int v = tid; v < CHUNK_F / 4; v += NB) {
    float4 t = *(const float4*)(gbase + v * 4);
    *(float4*)(lbase + v * 4) = t;
  }
#endif
}

// One block per (site i, j-half). Thread = batch element.
// partial[(half*NS + i)*NB + b] = sum_{j in half} J[i, j, s_bi, s_bj]
// J row slices stream HBM->LDS via async-to-LDS DMA, double-buffered.
__global__ __launch_bounds__(NB) void spin_partial_kernel(
    const float* __restrict__ J, const int* __restrict__ sT,
    float* __restrict__ partial) {
  __shared__ float lds[2][CHUNK_F];
  const int tid  = threadIdx.x;   // batch index b
  const int i    = blockIdx.x;    // site i
  const int half = blockIdx.y;    // j partition
  const int j0base = half * JPART;

  const float* Jrow = J + (size_t)i * (NS * PAIR) + (size_t)j0base * PAIR;
  const int a21 = sT[i * NB + tid] * NQ;   // s_i * 21, fixed for this block

  float acc = 0.0f;
  int cur = 0;

  copy_chunk(Jrow, &lds[0][0], tid);       // prefetch tile 0

  for (int k = 0; k < NCHUNK; ++k) {
#if USE_ASYNC
    __builtin_amdgcn_s_wait_asynccnt(0);   // my tile-k async copies landed
#endif
    __syncthreads();                       // everyone's tile-k data published;
                                           // also: all readers of buf cur^1 done
    if (k + 1 < NCHUNK)
      copy_chunk(Jrow + (size_t)(k + 1) * CHUNK_F, &lds[cur ^ 1][0], tid);

    const float* buf = &lds[cur][0];
    const int j0 = j0base + k * JC;

    // Stage the 16 state indices first (coalesced b32 loads, one wait),
    // then do the 16 LDS gathers.
    int c[JC];
#pragma unroll
    for (int jj = 0; jj < JC; ++jj) c[jj] = sT[(j0 + jj) * NB + tid];
#pragma unroll
    for (int jj = 0; jj < JC; ++jj) acc += buf[jj * PAIR + a21 + c[jj]];

    cur ^= 1;
  }

  partial[((size_t)half * NS + i) * NB + tid] = acc;
}

// sT[i*NB + b] = cfg[b*NS + i]  (coalesced writes)
__global__ __launch_bounds__(NB) void transpose_cfg(const int* __restrict__ cfg,
                                                    int* __restrict__ sT) {
  int i = blockIdx.x;
  int b = threadIdx.x;
  sT[i * NB + b] = cfg[b * NS + i];
}

// Deterministic fixed-order reduction + field term.
__global__ __launch_bounds__(NB) void finalize_kernel(
    const float* __restrict__ partial, const int* __restrict__ cfg,
    const float* __restrict__ field, float* __restrict__ out) {
  int b = threadIdx.x;
  float s = 0.0f;
  for (int w = 0; w < SPLIT * NS; ++w) s += partial[(size_t)w * NB + b];
  float fs = 0.0f;
  for (int i = 0; i < NS; ++i) {
    int c = cfg[b * NS + i];
    fs += field[i * NQ + c];
  }
  out[b] = 0.5f * s + fs;
}

extern "C" void kernel_launch(void* const* d_in, const int* in_sizes, int n_in,
                              void* d_out, int out_size, void* d_ws, size_t ws_size,
                              hipStream_t stream) {
  const int*   cfg   = (const int*)d_in[0];    // (256, 512) int32
  const float* J     = (const float*)d_in[1];  // (512, 512, 21, 21) f32
  const float* field = (const float*)d_in[2];  // (512, 21) f32
  float* out = (float*)d_out;                  // (256,) f32

  int*   sT      = (int*)d_ws;                                  // 512*256 i32 = 512 KB
  float* partial = (float*)((char*)d_ws + (size_t)NS * NB * 4); // 2*512*256 f32 = 1 MB

  transpose_cfg<<<NS, NB, 0, stream>>>(cfg, sT);

  dim3 grid(NS, SPLIT);
  spin_partial_kernel<<<grid, NB, 0, stream>>>(J, sT, partial);

  finalize_kernel<<<1, NB, 0, stream>>>(partial, cfg, field, out);
}